// USMSharp_3513283248863
// MI455X (gfx1250) — compile-verified
//
#include <hip/hip_runtime.h>
#include <math.h>

typedef float v2f __attribute__((ext_vector_type(2)));
typedef float v8f __attribute__((ext_vector_type(8)));

#define KS   51
#define RAD  25
#define BC   64          // output cols (c of pass input) per workgroup
#define BR   64          // output rows (r of pass input) per workgroup
#define PITCH 72         // LDS row pitch (dwords); 2*PITCH%64=16 -> conflict-free kh split
#define LROWS (BR + 52)  // 116: rows r0-25 .. r0-25+48+67
#define NTHREADS 512     // 16 waves; waves tiled 4(c) x 4(r)
#define NCHUNK 17        // ceil(66/4): banded K dimension in chunks of 4

// One pass: O[c][r] = sum_k g[k] * I[r+k-25][c]   (vertical conv, transposed store)
// Applying it twice gives the full separable 2D blur in original orientation.
// MODE 0: plain (out0 = conv result)
// MODE 2: epilogue computes blur1 -> out0, mask -> out1 (needs img)
// MODE 4: epilogue: soft_mask = conv result; final blend -> out0 (needs img, b1)
template <int MODE>
__global__ __launch_bounds__(NTHREADS) void usm_pass(
    const float* __restrict__ in,   // pass conv input, R x C row-major planes
    const float* __restrict__ g,    // 51-tap gaussian kernel
    const float* __restrict__ img,  // original image (MODE 2/4)
    const float* __restrict__ b1,   // blur1 (MODE 4)
    float* __restrict__ out0,       // primary output
    float* __restrict__ out1,       // mask output (MODE 2)
    int R, int C)
{
    __shared__ float tile[LROWS * PITCH];
    __shared__ float gsh[128];      // zero-guarded kernel: gsh[32+i] = g[i]

    const int tid = threadIdx.x;
    const int ch  = blockIdx.z;
    const int c0  = blockIdx.x * BC;
    const int r0  = blockIdx.y * BR;
    const size_t plane = (size_t)R * C;
    const float* I = in + (size_t)ch * plane;

    if (tid < 128) gsh[tid] = 0.0f;
    __syncthreads();
    if (tid < KS) gsh[32 + tid] = g[tid];

    // Cooperative load of the haloed input tile with reflect padding (rows only;
    // columns are always in range since C % BC == 0).
    for (int idx = tid; idx < LROWS * BC; idx += NTHREADS) {
        int lr = idx >> 6;      // /BC
        int lc = idx & (BC - 1);
        int rr = r0 - RAD + lr;
        rr = (rr < 0) ? -rr : rr;
        rr = (rr >= R) ? (2 * R - 2 - rr) : rr;
        tile[lr * PITCH + lc] = I[(size_t)rr * C + (c0 + lc)];
    }
    __syncthreads();

    const int wave = tid >> 5;       // 0..15
    const int lane = tid & 31;
    const int wc = wave & 3;         // 16-col sub-tile
    const int wr = wave >> 2;        // 16-row sub-tile
    const int n  = lane & 15;        // lane index within half (== m for A, == n for B)
    const int kh = lane >> 4;        // which K-pair this half-wave holds (kk = 2*kh + v)

    // Precompute banded-B registers: B_t[kk][n] = g[4t + kk - n] (zero outside band).
    float bv0[NCHUNK], bv1[NCHUNK];
#pragma unroll
    for (int t = 0; t < NCHUNK; ++t) {
        int j = 4 * t + 2 * kh;
        bv0[t] = gsh[32 + j - n];
        bv1[t] = gsh[33 + j - n];
    }

    // D[m][n] = sum_j tile[wr*16 + j][wc*16 + m] * g[j - n],  j = 0..67
    v8f acc = {0.f, 0.f, 0.f, 0.f, 0.f, 0.f, 0.f, 0.f};
    const int abase = (wr * 16) * PITCH + wc * 16 + n;   // A lane: m == n
#pragma unroll
    for (int t = 0; t < NCHUNK; ++t) {
        int ro = (4 * t + 2 * kh) * PITCH;
        v2f a, b;
        a.x = tile[abase + ro];
        a.y = tile[abase + ro + PITCH];
        b.x = bv0[t];
        b.y = bv1[t];
        acc = __builtin_amdgcn_wmma_f32_16x16x4_f32(
            false, a, false, b, (short)0, acc, false, false);
    }

    // D layout: vgpr v, lane l -> (M = v + 8*(l>=16), N = l%16)
    // O is C x R row-major: O[c0 + wc*16 + M][r0 + wr*16 + N]
    const int orow0 = c0 + wc * 16 + 8 * kh;
    const int ocol  = r0 + wr * 16 + n;
    const size_t obase = (size_t)ch * plane + (size_t)ocol;
#pragma unroll
    for (int v = 0; v < 8; ++v) {
        size_t o = obase + (size_t)(orow0 + v) * R;
        float s = acc[v];
        if (MODE == 0) {
            out0[o] = s;
        } else if (MODE == 2) {
            float ip = img[o];
            float m  = (fabsf(ip - s) * 255.0f > 10.0f) ? 1.0f : 0.0f;
            out0[o] = s;   // blur1
            out1[o] = m;   // mask
        } else {
            float ip  = img[o];
            float bb  = b1[o];
            float res = ip - bb;
            float sharp = fminf(fmaxf(ip + 0.5f * res, 0.0f), 1.0f);
            out0[o] = s * sharp + (1.0f - s) * ip;   // s == soft_mask
        }
    }
}

extern "C" void kernel_launch(void* const* d_in, const int* in_sizes, int n_in,
                              void* d_out, int out_size, void* d_ws, size_t ws_size,
                              hipStream_t stream) {
    const float* img = (const float*)d_in[0];
    const float* g   = (const float*)d_in[1];
    float* out = (float*)d_out;

    const int R = 512, C = 512, NCH = 16 * 3;
    const size_t plane_bytes = (size_t)R * C * sizeof(float);

    float* t    = (float*)d_ws;                                   // transposed temp
    float* mask = (float*)((char*)d_ws + (size_t)NCH * plane_bytes);

    dim3 grid(C / BC, R / BR, NCH);
    dim3 block(NTHREADS);

    // pass1: vblur(img) -> t (transposed)
    usm_pass<0><<<grid, block, 0, stream>>>(img, g, nullptr, nullptr, t, nullptr, R, C);
    // pass2: vblur(t) -> blur1 (original layout, stored in d_out) + mask -> ws
    usm_pass<2><<<grid, block, 0, stream>>>(t, g, img, nullptr, out, mask, R, C);
    // pass3: vblur(mask) -> t (transposed)
    usm_pass<0><<<grid, block, 0, stream>>>(mask, g, nullptr, nullptr, t, nullptr, R, C);
    // pass4: vblur(t) = soft_mask; fused final blend -> d_out (reads b1 from d_out first)
    usm_pass<4><<<grid, block, 0, stream>>>(t, g, img, out, out, nullptr, R, C);
}